// CrossAtt_17566416241415
// MI455X (gfx1250) — compile-verified
//
#include <hip/hip_runtime.h>
#include <hip/hip_bf16.h>
#include <stdint.h>

typedef __attribute__((ext_vector_type(2))) float v2f;
typedef __attribute__((ext_vector_type(8))) float v8f;

#define EPSV 1e-5f

__device__ __forceinline__ float hswish(float x) {
    float t = fminf(fmaxf(x + 3.f, 0.f), 6.f);
    return x * t * (1.f / 6.f);
}

// Async global->LDS copy (CDNA5): writes LDS directly, tracked by ASYNCcnt.
// lds_off = wave-relative LDS byte offset (low 32 bits of a generic LDS pointer).
__device__ __forceinline__ void async_ld_b32(unsigned lds_off, const float* gp) {
    asm volatile("global_load_async_to_lds_b32 %0, %1, off"
                 :: "v"(lds_off), "v"(gp) : "memory");
}
__device__ __forceinline__ void wait_asynccnt0() {
    asm volatile("s_wait_asynccnt 0x0" ::: "memory");
}

// ---------------------------------------------------------------------------
// Generic batched GEMM on V_WMMA_F32_16X16X4_F32 with double-buffered
// async-to-LDS staging.
//   C[z][m][n] = epilogue( sum_k A[z][m][k] * B[z][k][n] )
// transA: A element at A[gk*lda+gm] (column-major source, used for k^T v)
// transB: B element at B[gn*ldb+gk] (weight stored row-major (N,K))
// Batch offset for z: (z/zInner)*Zo + (z%zInner)*Zi  (covers (b,h) double index)
// Epilogues: 0=store, 1=+bias, 2=BN(scale,bias)+hardswish, 3=*alpha
// Block: 256 threads = 8 waves; tile 128(M) x 64(N); K-chunk 16.
// NOTE: all launches below are geometry-checked so unguarded tile loads stay
// inside allocated buffers; stores remain guarded.
// ---------------------------------------------------------------------------
__global__ __launch_bounds__(256)
void wmma_gemm_f32(const float* __restrict__ A, const float* __restrict__ Bm,
                   float* __restrict__ C,
                   int M, int Nd, int K,
                   long long lda, long long ldb, long long ldc,
                   int transA, int transB,
                   long long aZo, long long aZi,
                   long long bZo, long long bZi,
                   long long cZo, long long cZi, int zInner,
                   int ep, const float* __restrict__ sc,
                   const float* __restrict__ bi, float alpha) {
    __shared__ float As[2][16][128];
    __shared__ float Bs[2][16][64];

    const int z = blockIdx.z;
    const long long aOff = (long long)(z / zInner) * aZo + (long long)(z % zInner) * aZi;
    const long long bOff = (long long)(z / zInner) * bZo + (long long)(z % zInner) * bZi;
    const long long cOff = (long long)(z / zInner) * cZo + (long long)(z % zInner) * cZi;

    const int tid  = threadIdx.x;
    const int wave = tid >> 5;
    const int lane = tid & 31;
    const int m0 = blockIdx.x * 128;
    const int n0 = blockIdx.y * 64;

    const unsigned asBase = (unsigned)(uintptr_t)(&As[0][0][0]);
    const unsigned bsBase = (unsigned)(uintptr_t)(&Bs[0][0][0]);

    const float* Ab = A + aOff;
    const float* Bb = Bm + bOff;

    // Issue the async loads for one K-chunk into LDS buffer `buf`.
    auto stage = [&](int buf, int k0) {
        // A tile 128x16, k-major in LDS: As[buf][kk][mm]
#pragma unroll
        for (int j = 0; j < 8; ++j) {
            int i = tid + 256 * j;
            int kk = i & 15, mm = i >> 4;
            long long gm = m0 + mm, gk = k0 + kk;
            const float* gp = transA ? (Ab + gk * lda + gm) : (Ab + gm * lda + gk);
            async_ld_b32(asBase + (unsigned)((buf * 2048 + kk * 128 + mm) * 4), gp);
        }
        // B tile 16x64: Bs[buf][kk][nn]
#pragma unroll
        for (int j = 0; j < 4; ++j) {
            int i = tid + 256 * j;
            int kk = i & 15, nn = i >> 4;
            long long gn = n0 + nn, gk = k0 + kk;
            const float* gp = transB ? (Bb + gn * ldb + gk) : (Bb + gk * ldb + gn);
            async_ld_b32(bsBase + (unsigned)((buf * 1024 + kk * 64 + nn) * 4), gp);
        }
    };

    v8f acc[4] = {};

    const int mlane = lane & 15;
    const int khalf = (lane >> 4) << 1;   // 0 for lanes 0-15, 2 for lanes 16-31

    stage(0, 0);
    int cur = 0;
    for (int k0 = 0; k0 < K; k0 += 16) {
        wait_asynccnt0();      // own async loads complete (LDS data written)
        __syncthreads();       // all waves' loads complete; prior reads done
        if (k0 + 16 < K) stage(cur ^ 1, k0 + 16);   // overlap next chunk with WMMA

#pragma unroll
        for (int kk = 0; kk < 16; kk += 4) {
            v2f af;
            af.x = As[cur][kk + khalf + 0][wave * 16 + mlane];
            af.y = As[cur][kk + khalf + 1][wave * 16 + mlane];
#pragma unroll
            for (int nt = 0; nt < 4; nt++) {
                v2f bf;
                bf.x = Bs[cur][kk + khalf + 0][nt * 16 + mlane];
                bf.y = Bs[cur][kk + khalf + 1][nt * 16 + mlane];
                acc[nt] = __builtin_amdgcn_wmma_f32_16x16x4_f32(
                    false, af, false, bf, (short)0, acc[nt], false, false);
            }
        }
        cur ^= 1;
    }

    // Epilogue. C/D layout: VGPR i -> row (wave*16 + i + (lane>=16 ? 8:0)), col lane%16.
    const int mh = (lane >> 4) * 8;
#pragma unroll
    for (int nt = 0; nt < 4; nt++) {
        int gn = n0 + nt * 16 + mlane;
        if (gn >= Nd) continue;
        float s = 1.f, b2 = 0.f;
        if (ep == 1) b2 = bi[gn];
        else if (ep == 2) { s = sc[gn] * rsqrtf(1.f + EPSV); b2 = bi[gn]; }
#pragma unroll
        for (int i = 0; i < 8; i++) {
            int gm = m0 + wave * 16 + mh + i;
            if (gm >= M) continue;
            float v = acc[nt][i];
            if (ep == 1) v += b2;
            else if (ep == 2) v = hswish(v * s + b2);
            else if (ep == 3) v *= alpha;
            C[cOff + (long long)gm * ldc + gn] = v;
        }
    }
}

// ---------------------------------------------------------------------------
// Depthwise conv on token-major image data (64x64 image, 128 channels).
// in: (B_, 4096, inLd) with channel window pre-offset into pointer.
// out: (B_, 4096, 128) token-major (ready as GEMM A-matrix).
// grid (4096, B_), block 128 (thread = channel -> coalesced).
// ---------------------------------------------------------------------------
__global__ void dwconv_tok(const float* __restrict__ in, const float* __restrict__ w,
                           float* __restrict__ out, int ksz, int pad, long long inLd) {
    int c = threadIdx.x;
    int n = blockIdx.x;
    int b = blockIdx.y;
    int yy = n >> 6, xx = n & 63;
    const float* wp = w + c * ksz * ksz;
    const float* ip = in + (long long)b * 4096 * inLd + c;
    float acc = 0.f;
    for (int i = 0; i < ksz; i++) {
        int sy = yy + i - pad;
        if ((unsigned)sy >= 64u) continue;
        for (int j = 0; j < ksz; j++) {
            int sx = xx + j - pad;
            if ((unsigned)sx >= 64u) continue;
            acc += wp[i * ksz + j] * ip[(long long)(sy * 64 + sx) * inLd];
        }
    }
    out[((long long)b * 4096 + n) * 128 + c] = acc;
}

// Segment-0: eval BatchNorm + hardswish, elementwise. grid = B_*4096, block 128.
__global__ void bn_hswish_ew(const float* __restrict__ in, const float* __restrict__ g,
                             const float* __restrict__ bb, float* __restrict__ out,
                             long long inLd, long long outLd) {
    long long t = blockIdx.x;
    int c = threadIdx.x;
    float s = g[c] * rsqrtf(1.f + EPSV);
    out[t * outLd + c] = hswish(in[t * inLd + c] * s + bb[c]);
}

// LayerNorm(128) + hardswish for the "loc" branch. grid = tokens, block 128.
__global__ void ln_hswish_k(const float* __restrict__ in, const float* __restrict__ g,
                            const float* __restrict__ bb, float* __restrict__ out) {
    __shared__ float red[128];
    long long t = blockIdx.x;
    int c = threadIdx.x;
    float v = in[t * 128 + c];
    red[c] = v; __syncthreads();
    for (int s = 64; s > 0; s >>= 1) { if (c < s) red[c] += red[c + s]; __syncthreads(); }
    float m = red[0] * (1.f / 128.f);
    __syncthreads();
    float d = v - m;
    red[c] = d * d; __syncthreads();
    for (int s = 64; s > 0; s >>= 1) { if (c < s) red[c] += red[c + s]; __syncthreads(); }
    float var = red[0] * (1.f / 128.f);
    out[t * 128 + c] = hswish(d * rsqrtf(var + EPSV) * g[c] + bb[c]);
}

// Column-softmax stats over tokens: k (8,4096,512) -> max,sumexp per (b,c).
// grid 8, block 512 (thread = channel, coalesced across c).
__global__ void softmax_colstats(const float* __restrict__ k, float* __restrict__ cmax,
                                 float* __restrict__ csum) {
    int c = threadIdx.x;
    int b = blockIdx.x;
    const float* p = k + (long long)b * 4096 * 512 + c;
    float mx = -3.0e38f;
    for (int n = 0; n < 4096; n++) mx = fmaxf(mx, p[(long long)n * 512]);
    float sm = 0.f;
    for (int n = 0; n < 4096; n++) sm += __expf(p[(long long)n * 512] - mx);
    cmax[b * 512 + c] = mx;
    csum[b * 512 + c] = sm;
}

// Apply softmax. grid = 8*4096 tokens, block 512.
__global__ void softmax_apply(const float* __restrict__ k, const float* __restrict__ cmax,
                              const float* __restrict__ csum, float* __restrict__ ks) {
    long long t = blockIdx.x;
    int c = threadIdx.x;
    int b = (int)(t >> 12);
    ks[t * 512 + c] = __expf(k[t * 512 + c] - cmax[b * 512 + c]) / csum[b * 512 + c];
}

// CRPE: o[b,n,c] += q[b,n,c] * dwconv(v)[b,c,n] with per-range kernel size.
// grid (4096, 4, 8), block 128 -> c = 0..511.
__global__ void crpe_add(const float* __restrict__ v, const float* __restrict__ q,
                         const float* __restrict__ w3, const float* __restrict__ w5,
                         const float* __restrict__ w7, float* __restrict__ o) {
    int c = threadIdx.x + blockIdx.y * 128;
    int n = blockIdx.x;
    int b = blockIdx.z;
    int ksz, pad;
    const float* wp;
    if (c < 128)      { ksz = 3; pad = 1; wp = w3 + c * 9; }
    else if (c < 320) { ksz = 5; pad = 2; wp = w5 + (c - 128) * 25; }
    else              { ksz = 7; pad = 3; wp = w7 + (c - 320) * 49; }
    int yy = n >> 6, xx = n & 63;
    const float* vp = v + (long long)b * 4096 * 512 + c;
    float acc = 0.f;
    for (int i = 0; i < ksz; i++) {
        int sy = yy + i - pad;
        if ((unsigned)sy >= 64u) continue;
        for (int j = 0; j < ksz; j++) {
            int sx = xx + j - pad;
            if ((unsigned)sx >= 64u) continue;
            acc += wp[i * ksz + j] * vp[(long long)(sy * 64 + sx) * 512];
        }
    }
    long long t = (long long)b * 4096 + n;
    o[t * 640 + c] += q[t * 512 + c] * acc;
}

// Last 128 output channels: kv_loc + q_loc. grid = 8*4096, block 128.
__global__ void loc_sum(const float* __restrict__ lv, const float* __restrict__ lq,
                        float* __restrict__ o) {
    long long t = blockIdx.x;
    int c = threadIdx.x;
    o[t * 640 + 512 + c] = lv[t * 128 + c] + lq[t * 128 + c];
}

extern "C" void kernel_launch(void* const* d_in, const int* in_sizes, int n_in,
                              void* d_out, int out_size, void* d_ws, size_t ws_size,
                              hipStream_t stream) {
    const float* x       = (const float*)d_in[0];
    const float* y       = (const float*)d_in[1];
    const float* kv_w    = (const float*)d_in[2];
    const float* proj_w  = (const float*)d_in[3];
    const float* proj_b  = (const float*)d_in[4];
    const float* crpe_w3 = (const float*)d_in[5];
    const float* crpe_w5 = (const float*)d_in[6];
    const float* crpe_w7 = (const float*)d_in[7];
    const float* aq_dw3  = (const float*)d_in[8];
    const float* aq_dw5  = (const float*)d_in[9];
    const float* aq_dw7  = (const float*)d_in[10];
    const float* aq_pw   = (const float*)d_in[11];
    const float* aq_ln_g = (const float*)d_in[12];
    const float* aq_ln_b = (const float*)d_in[13];
    const float* aq_bn_g = (const float*)d_in[14];
    const float* aq_bn_b = (const float*)d_in[15];
    const float* akv_dw3 = (const float*)d_in[16];
    const float* akv_dw5 = (const float*)d_in[17];
    const float* akv_dw7 = (const float*)d_in[18];
    const float* akv_pw  = (const float*)d_in[19];
    const float* akv_ln_g= (const float*)d_in[20];
    const float* akv_ln_b= (const float*)d_in[21];
    const float* akv_bn_g= (const float*)d_in[22];
    const float* akv_bn_b= (const float*)d_in[23];
    float* out = (float*)d_out;

    // Workspace layout (fp32 elements)
    float* Wp     = (float*)d_ws;
    float* kvbuf  = Wp;                       // (2,8,4096,640) = 41,943,040
    float* aggkv  = kvbuf  + 41943040LL;      // (16,4096,512)  = 33,554,432
    float* aggq   = aggkv  + 33554432LL;      // (8,4096,512)   = 16,777,216
    float* ksmb   = aggq   + 16777216LL;      // (8,4096,512)
    float* dwtmp  = ksmb   + 16777216LL;      // (16,4096,128)  =  8,388,608
    float* locv   = dwtmp  +  8388608LL;      // (8,4096,128)   =  4,194,304
    float* locq   = locv   +  4194304LL;
    float* locraw = locq   +  4194304LL;
    float* obuf   = locraw +  4194304LL;      // (8,4096,640)   = 20,971,520
    float* cmax   = obuf   + 20971520LL;      // (8,512)
    float* csum   = cmax   +  4096LL;
    float* ktv    = csum   +  4096LL;         // (64,64,64)

    const float SCALE = 0.11180339887498949f; // (640/8)^-0.5
    dim3 blk(256);

    // 1) kv projection: k half and v half (M=32768, N=640, K=640 each)
    wmma_gemm_f32<<<dim3(256, 10, 1), blk, 0, stream>>>(
        x, kv_w, kvbuf, 32768, 640, 640, 640, 640, 640, 0, 1,
        0, 0, 0, 0, 0, 0, 1, 0, nullptr, nullptr, 1.f);
    wmma_gemm_f32<<<dim3(256, 10, 1), blk, 0, stream>>>(
        x, kv_w + 640LL * 640, kvbuf + 20971520LL, 32768, 640, 640, 640, 640, 640, 0, 1,
        0, 0, 0, 0, 0, 0, 1, 0, nullptr, nullptr, 1.f);

    // 2) Aggregator on kv (B_=16)
    bn_hswish_ew<<<65536, 128, 0, stream>>>(kvbuf, akv_bn_g, akv_bn_b, aggkv, 640, 512);
    // seg1: dw3[1] + pw[1] + bn[1]
    dwconv_tok<<<dim3(4096, 16), 128, 0, stream>>>(kvbuf + 128, akv_dw3 + 1152, dwtmp, 3, 1, 640);
    wmma_gemm_f32<<<dim3(512, 2, 1), blk, 0, stream>>>(
        dwtmp, akv_pw + 16384, aggkv + 128, 65536, 128, 128, 128, 128, 512, 0, 1,
        0, 0, 0, 0, 0, 0, 1, 2, akv_bn_g + 128, akv_bn_b + 128, 1.f);
    // seg2: dw5 + pw[2] + bn[2]
    dwconv_tok<<<dim3(4096, 16), 128, 0, stream>>>(kvbuf + 256, akv_dw5, dwtmp, 5, 2, 640);
    wmma_gemm_f32<<<dim3(512, 2, 1), blk, 0, stream>>>(
        dwtmp, akv_pw + 2 * 16384, aggkv + 256, 65536, 128, 128, 128, 128, 512, 0, 1,
        0, 0, 0, 0, 0, 0, 1, 2, akv_bn_g + 256, akv_bn_b + 256, 1.f);
    // seg3: dw7 + pw[3] + bn[3]
    dwconv_tok<<<dim3(4096, 16), 128, 0, stream>>>(kvbuf + 384, akv_dw7, dwtmp, 7, 3, 640);
    wmma_gemm_f32<<<dim3(512, 2, 1), blk, 0, stream>>>(
        dwtmp, akv_pw + 3 * 16384, aggkv + 384, 65536, 128, 128, 128, 128, 512, 0, 1,
        0, 0, 0, 0, 0, 0, 1, 2, akv_bn_g + 384, akv_bn_b + 384, 1.f);
    // loc branch (v half only): dw3[0] + pw[0] -> LN -> hardswish
    dwconv_tok<<<dim3(4096, 8), 128, 0, stream>>>(
        kvbuf + 8LL * 4096 * 640 + 512, akv_dw3, dwtmp, 3, 1, 640);
    wmma_gemm_f32<<<dim3(256, 2, 1), blk, 0, stream>>>(
        dwtmp, akv_pw, locraw, 32768, 128, 128, 128, 128, 128, 0, 1,
        0, 0, 0, 0, 0, 0, 1, 0, nullptr, nullptr, 1.f);
    ln_hswish_k<<<32768, 128, 0, stream>>>(locraw, akv_ln_g, akv_ln_b, locv);

    // 3) Aggregator on q = y (B_=8)
    bn_hswish_ew<<<32768, 128, 0, stream>>>(y, aq_bn_g, aq_bn_b, aggq, 640, 512);
    dwconv_tok<<<dim3(4096, 8), 128, 0, stream>>>(y + 128, aq_dw3 + 1152, dwtmp, 3, 1, 640);
    wmma_gemm_f32<<<dim3(256, 2, 1), blk, 0, stream>>>(
        dwtmp, aq_pw + 16384, aggq + 128, 32768, 128, 128, 128, 128, 512, 0, 1,
        0, 0, 0, 0, 0, 0, 1, 2, aq_bn_g + 128, aq_bn_b + 128, 1.f);
    dwconv_tok<<<dim3(4096, 8), 128, 0, stream>>>(y + 256, aq_dw5, dwtmp, 5, 2, 640);
    wmma_gemm_f32<<<dim3(256, 2, 1), blk, 0, stream>>>(
        dwtmp, aq_pw + 2 * 16384, aggq + 256, 32768, 128, 128, 128, 128, 512, 0, 1,
        0, 0, 0, 0, 0, 0, 1, 2, aq_bn_g + 256, aq_bn_b + 256, 1.f);
    dwconv_tok<<<dim3(4096, 8), 128, 0, stream>>>(y + 384, aq_dw7, dwtmp, 7, 3, 640);
    wmma_gemm_f32<<<dim3(256, 2, 1), blk, 0, stream>>>(
        dwtmp, aq_pw + 3 * 16384, aggq + 384, 32768, 128, 128, 128, 128, 512, 0, 1,
        0, 0, 0, 0, 0, 0, 1, 2, aq_bn_g + 384, aq_bn_b + 384, 1.f);
    dwconv_tok<<<dim3(4096, 8), 128, 0, stream>>>(y + 512, aq_dw3, dwtmp, 3, 1, 640);
    wmma_gemm_f32<<<dim3(256, 2, 1), blk, 0, stream>>>(
        dwtmp, aq_pw, locraw, 32768, 128, 128, 128, 128, 128, 0, 1,
        0, 0, 0, 0, 0, 0, 1, 0, nullptr, nullptr, 1.f);
    ln_hswish_k<<<32768, 128, 0, stream>>>(locraw, aq_ln_g, aq_ln_b, locq);

    // 4) Token-axis softmax of k (aggkv batches 0..7)
    softmax_colstats<<<8, 512, 0, stream>>>(aggkv, cmax, csum);
    softmax_apply<<<32768, 512, 0, stream>>>(aggkv, cmax, csum, ksmb);

    // 5) kTv = k_sm^T @ v : per (b,h), M=64,N=64,K=4096 (transA)
    wmma_gemm_f32<<<dim3(1, 1, 64), blk, 0, stream>>>(
        ksmb, aggkv + 8LL * 4096 * 512, ktv, 64, 64, 4096, 512, 512, 64, 1, 0,
        4096LL * 512, 64, 4096LL * 512, 64, 32768, 4096, 8,
        0, nullptr, nullptr, 1.f);

    // 6) eff = q @ kTv, scaled: per (b,h), M=4096,N=64,K=64 -> obuf channels 0..511
    wmma_gemm_f32<<<dim3(32, 4, 64), blk, 0, stream>>>(
        aggq, ktv, obuf, 4096, 64, 64, 512, 64, 640, 0, 0,
        4096LL * 512, 64, 32768, 4096, 4096LL * 640, 64, 8,
        3, nullptr, nullptr, SCALE);

    // 7) CRPE: obuf += q * dwconv(v)
    crpe_add<<<dim3(4096, 4, 8), 128, 0, stream>>>(
        aggkv + 8LL * 4096 * 512, aggq, crpe_w3, crpe_w5, crpe_w7, obuf);

    // 8) loc channels
    loc_sum<<<32768, 128, 0, stream>>>(locv, locq, obuf);

    // 9) final projection + bias: M=32768, N=640, K=640
    wmma_gemm_f32<<<dim3(256, 10, 1), blk, 0, stream>>>(
        obuf, proj_w, out, 32768, 640, 640, 640, 640, 640, 0, 1,
        0, 0, 0, 0, 0, 0, 1, 1, nullptr, proj_b, 1.f);
}